// StatementClassfier_59691455480552
// MI455X (gfx1250) — compile-verified
//
#include <hip/hip_runtime.h>
#include <hip/hip_bf16.h>
#include <math.h>

typedef __attribute__((ext_vector_type(2))) float v2f;
typedef __attribute__((ext_vector_type(8))) float v8f;

#define DH 32
#define NEG_SLOPE 0.2f

// ---------- ordered-uint mapping for float atomic max ----------
__device__ __forceinline__ unsigned f2ord(float f) {
    unsigned b = __float_as_uint(f);
    return (b & 0x80000000u) ? ~b : (b | 0x80000000u);
}
__device__ __forceinline__ float ord2f(unsigned o) {
    return __uint_as_float((o & 0x80000000u) ? (o & 0x7fffffffu) : ~o);
}

// ---------- H[r] = X @ W[r], fp32 WMMA 16x16x4, one wave per 16-node tile ----------
// X: [N][din], W: [R=2][din][32], H: [2][N][32]
__global__ __launch_bounds__(256) void transform_wmma(
    const float* __restrict__ X, const float* __restrict__ W,
    float* __restrict__ H, int N, int din)
{
    int wave  = (blockIdx.x * blockDim.x + threadIdx.x) >> 5;
    int lane  = threadIdx.x & 31;
    int node0 = wave * 16;
    if (node0 >= N) return;                 // wave-uniform exit (EXEC stays full)

    int m     = lane & 15;
    int khalf = (lane >> 4) << 1;           // 0 for lanes 0-15, 2 for lanes 16-31

    v8f acc[2][2] = {};                     // [relation][16-col tile]
    const float* xrow = X + (size_t)(node0 + m) * din;

    for (int k = 0; k < din; k += 4) {
        v2f a;
        a.x = xrow[k + khalf];
        a.y = xrow[k + khalf + 1];
        #pragma unroll
        for (int r = 0; r < 2; ++r) {
            const float* wbase = W + ((size_t)r * din + (k + khalf)) * DH;
            #pragma unroll
            for (int t = 0; t < 2; ++t) {
                v2f b;
                b.x = wbase[t * 16 + m];        // row k+khalf
                b.y = wbase[DH + t * 16 + m];   // row k+khalf+1
                acc[r][t] = __builtin_amdgcn_wmma_f32_16x16x4_f32(
                    false, a, false, b, (short)0, acc[r][t], false, false);
            }
        }
    }
    // C/D layout: VGPR j -> row M=j (lanes 0-15) / M=j+8 (lanes 16-31), col = lane&15
    int mrow = (lane >> 4) << 3;
    #pragma unroll
    for (int r = 0; r < 2; ++r)
        #pragma unroll
        for (int t = 0; t < 2; ++t)
            #pragma unroll
            for (int j = 0; j < 8; ++j) {
                int node = node0 + mrow + j;
                H[((size_t)r * N + node) * DH + t * 16 + m] = acc[r][t][j];
            }
}

// ---------- per (relation,node): qn = H.Q, kn = H.K ----------
__global__ void qk_kernel(const float* __restrict__ H,
                          const float* __restrict__ Q, const float* __restrict__ K,
                          float* __restrict__ qn, float* __restrict__ kn, int N)
{
    int t = blockIdx.x * blockDim.x + threadIdx.x;   // t in [0, 2N)
    if (t >= 2 * N) return;
    int r = t / N;
    const float* hrow = H + (size_t)t * DH;
    const float* qv = Q + r * DH;
    const float* kv = K + r * DH;
    float q = 0.f, k = 0.f;
    #pragma unroll
    for (int f = 0; f < DH; ++f) { float hv = hrow[f]; q += hv * qv[f]; k += hv * kv[f]; }
    qn[t] = q; kn[t] = k;
}

// ---------- per edge: leaky_relu(q[dst]+k[src]), segment max ----------
__global__ void score_kernel(const float* __restrict__ qn, const float* __restrict__ kn,
                             const int* __restrict__ src, const int* __restrict__ dst,
                             const int* __restrict__ et,
                             float* __restrict__ sc, unsigned* __restrict__ maxb,
                             int N, int E)
{
    int e = blockIdx.x * blockDim.x + threadIdx.x;
    if (e >= E) return;
    int r = et[e], d = dst[e], s = src[e];
    float v = qn[(size_t)r * N + d] + kn[(size_t)r * N + s];
    v = v > 0.f ? v : NEG_SLOPE * v;
    sc[e] = v;
    atomicMax(&maxb[d], f2ord(v));
}

// ---------- per edge: exp(e - max[dst]), segment sum (sc -> ex in place) ----------
__global__ void exp_kernel(float* __restrict__ sc, const unsigned* __restrict__ maxb,
                           const int* __restrict__ dst, float* __restrict__ sumb, int E)
{
    int e = blockIdx.x * blockDim.x + threadIdx.x;
    if (e >= E) return;
    int d = dst[e];
    float ex = expf(sc[e] - ord2f(maxb[d]));
    sc[e] = ex;
    atomicAdd(&sumb[d], ex);
}

// ---------- one warp per edge: acc[dst] += alpha * H[r][src] (lane = feature) ----------
__global__ __launch_bounds__(256) void aggregate_kernel(
    const float* __restrict__ H, const int* __restrict__ src,
    const int* __restrict__ dst, const int* __restrict__ et,
    const float* __restrict__ ex, const float* __restrict__ sumb,
    float* __restrict__ acc, int N, int E)
{
    int lane = threadIdx.x & 31;
    int e = (blockIdx.x * blockDim.x + threadIdx.x) >> 5;
    if (e >= E) return;
    int d = dst[e], s = src[e], r = et[e];
    float alpha = ex[e] / (sumb[d] + 1e-16f);
    float v = alpha * H[((size_t)r * N + s) * DH + lane];
    atomicAdd(&acc[(size_t)d * DH + lane], v);
}

// ---------- h = relu(acc + bias), accumulate column sum / sumsq ----------
__global__ __launch_bounds__(256) void bn_stats(float* __restrict__ h,
                                                const float* __restrict__ bias,
                                                float* __restrict__ colsum,
                                                float* __restrict__ colsq, int N)
{
    __shared__ float ssum[256], ssq[256];
    int f = threadIdx.x & 31;
    int grp = threadIdx.x >> 5;
    float b = bias[f];
    float lsum = 0.f, lsq = 0.f;
    for (int n = blockIdx.x * 8 + grp; n < N; n += gridDim.x * 8) {
        size_t idx = (size_t)n * DH + f;
        float v = h[idx] + b;
        v = v > 0.f ? v : 0.f;
        h[idx] = v;
        lsum += v; lsq += v * v;
    }
    ssum[threadIdx.x] = lsum; ssq[threadIdx.x] = lsq;
    __syncthreads();
    for (int stride = 128; stride >= 32; stride >>= 1) {
        if (threadIdx.x < (unsigned)stride) {
            ssum[threadIdx.x] += ssum[threadIdx.x + stride];
            ssq[threadIdx.x]  += ssq[threadIdx.x + stride];
        }
        __syncthreads();
    }
    if (threadIdx.x < 32) {
        atomicAdd(&colsum[f], ssum[threadIdx.x]);
        atomicAdd(&colsq[f],  ssq[threadIdx.x]);
    }
}

// ---------- h = g*(h-mu)*rsqrt(var+eps)+be ----------
__global__ void bn_apply(float* __restrict__ h, const float* __restrict__ colsum,
                         const float* __restrict__ colsq, const float* __restrict__ g,
                         const float* __restrict__ be, int N)
{
    int t = blockIdx.x * blockDim.x + threadIdx.x;
    if (t >= N * DH) return;
    int f = t & 31;
    float invN = 1.f / (float)N;
    float mu = colsum[f] * invN;
    float var = colsq[f] * invN - mu * mu;
    h[t] = g[f] * (h[t] - mu) * rsqrtf(var + 1e-5f) + be[f];
}

// ---------- MLP head: sigmoid(h@mw1+mb1)@mw2+mb2 ----------
__global__ void head_kernel(const float* __restrict__ h,
                            const float* __restrict__ mw1, const float* __restrict__ mb1,
                            const float* __restrict__ mw2, const float* __restrict__ mb2,
                            float* __restrict__ out, int N)
{
    int n = blockIdx.x * blockDim.x + threadIdx.x;
    if (n >= N) return;
    float hv[DH];
    #pragma unroll
    for (int i = 0; i < DH; ++i) hv[i] = h[(size_t)n * DH + i];
    float z[DH];
    #pragma unroll 4
    for (int j = 0; j < DH; ++j) {
        float a = mb1[j];
        #pragma unroll
        for (int i = 0; i < DH; ++i) a += hv[i] * mw1[i * DH + j];
        z[j] = 1.f / (1.f + expf(-a));
    }
    float o0 = mb2[0], o1 = mb2[1];
    #pragma unroll
    for (int j = 0; j < DH; ++j) { o0 += z[j] * mw2[j * 2]; o1 += z[j] * mw2[j * 2 + 1]; }
    out[n * 2 + 0] = o0;
    out[n * 2 + 1] = o1;
}

extern "C" void kernel_launch(void* const* d_in, const int* in_sizes, int n_in,
                              void* d_out, int out_size, void* d_ws, size_t ws_size,
                              hipStream_t stream)
{
    const int D_IN = 128;
    const int N = in_sizes[0] / D_IN;     // 100000
    const int E = in_sizes[2];            // 1600000

    const float* x    = (const float*)d_in[0];
    const int*   eidx = (const int*)d_in[1];
    const int*   etyp = (const int*)d_in[2];
    const int*   src  = eidx;
    const int*   dst  = eidx + E;

    const float* w[3]  = {(const float*)d_in[3],  (const float*)d_in[9],  (const float*)d_in[15]};
    const float* aq[3] = {(const float*)d_in[4],  (const float*)d_in[10], (const float*)d_in[16]};
    const float* ak[3] = {(const float*)d_in[5],  (const float*)d_in[11], (const float*)d_in[17]};
    const float* bb[3] = {(const float*)d_in[6],  (const float*)d_in[12], (const float*)d_in[18]};
    const float* gg[3] = {(const float*)d_in[7],  (const float*)d_in[13], (const float*)d_in[19]};
    const float* be[3] = {(const float*)d_in[8],  (const float*)d_in[14], (const float*)d_in[20]};
    const float* mw1 = (const float*)d_in[21];
    const float* mb1 = (const float*)d_in[22];
    const float* mw2 = (const float*)d_in[23];
    const float* mb2 = (const float*)d_in[24];

    // workspace carve-up (floats)
    float*    ws     = (float*)d_ws;
    float*    H      = ws;                         // 2*N*32
    float*    qn     = H    + (size_t)2 * N * DH;  // 2*N
    float*    kn     = qn   + (size_t)2 * N;       // 2*N
    float*    sc     = kn   + (size_t)2 * N;       // E (scores, then exp in place)
    unsigned* maxb   = (unsigned*)(sc + E);        // N
    float*    sumb   = (float*)(maxb + N);         // N
    float*    colsum = sumb + N;                   // 32
    float*    colsq  = colsum + DH;                // 32
    float*    accA   = colsq + DH;                 // N*32
    float*    accB   = accA + (size_t)N * DH;      // N*32

    const float* lin[3]  = {x, accA, accB};
    float*       lout[3] = {accA, accB, accA};
    const int    dins[3] = {D_IN, DH, DH};

    const int thr = 256;
    for (int l = 0; l < 3; ++l) {
        int waves  = N / 16;
        int tblk   = (waves * 32 + thr - 1) / thr;
        transform_wmma<<<tblk, thr, 0, stream>>>(lin[l], w[l], H, N, dins[l]);
        qk_kernel<<<(2 * N + thr - 1) / thr, thr, 0, stream>>>(H, aq[l], ak[l], qn, kn, N);

        hipMemsetAsync(maxb,    0, (size_t)N * sizeof(unsigned), stream);
        hipMemsetAsync(sumb,    0, (size_t)N * sizeof(float), stream);
        hipMemsetAsync(colsum,  0, 2 * DH * sizeof(float), stream);   // colsum+colsq
        hipMemsetAsync(lout[l], 0, (size_t)N * DH * sizeof(float), stream);

        score_kernel<<<(E + thr - 1) / thr, thr, 0, stream>>>(qn, kn, src, dst, etyp, sc, maxb, N, E);
        exp_kernel<<<(E + thr - 1) / thr, thr, 0, stream>>>(sc, maxb, dst, sumb, E);
        aggregate_kernel<<<(E * 32 + thr - 1) / thr, thr, 0, stream>>>(H, src, dst, etyp, sc, sumb, lout[l], N, E);

        bn_stats<<<440, thr, 0, stream>>>(lout[l], bb[l], colsum, colsq, N);
        bn_apply<<<(N * DH + thr - 1) / thr, thr, 0, stream>>>(lout[l], colsum, colsq, gg[l], be[l], N);
    }
    head_kernel<<<(N + thr - 1) / thr, thr, 0, stream>>>(accA, mw1, mb1, mw2, mb2, (float*)d_out, N);
}